// Top2Router_38508676776576
// MI455X (gfx1250) — compile-verified
//
#include <hip/hip_runtime.h>
#include <hip/hip_bf16.h>
#include <math.h>

typedef float v2f __attribute__((ext_vector_type(2)));
typedef float v4f __attribute__((ext_vector_type(4)));
typedef float v8f __attribute__((ext_vector_type(8)));

#define S_TOKENS 8192
#define E_EXPERTS 64
#define CAPACITY 320
#define SEC ((size_t)S_TOKENS * E_EXPERTS * CAPACITY) // 167,772,160 elements

// workspace layout (4-byte words). Total ~198 KB.
#define OFF_CNT1 0
#define OFF_CNT2 64
#define OFF_ME   128
#define OFF_IDX1 256
#define OFF_IDX2 (OFF_IDX1 + S_TOKENS)
#define OFF_W1   (OFF_IDX2 + S_TOKENS)
#define OFF_W2   (OFF_W1 + S_TOKENS)
#define OFF_MX   (OFF_W2 + S_TOKENS)
#define OFF_INV  (OFF_MX + S_TOKENS)

// ---------------------------------------------------------------------------
// K0: zero-fill combine_weights + sec_mask (1.34 GB) with 16B NT stores,
//     and zero the workspace counters (cnt1/cnt2/me).
// ---------------------------------------------------------------------------
__global__ void k0_zero(float* __restrict__ out, unsigned* __restrict__ wsw) {
    const size_t total4 = (2 * SEC) / 4; // divisible by 4
    size_t tid    = (size_t)blockIdx.x * blockDim.x + threadIdx.x;
    size_t stride = (size_t)gridDim.x * blockDim.x;
    v4f z = {0.f, 0.f, 0.f, 0.f};
    v4f* o = (v4f*)out;
    for (size_t i = tid; i < total4; i += stride)
        __builtin_nontemporal_store(z, o + i);
    if (blockIdx.x == 0 && threadIdx.x < 256) wsw[threadIdx.x] = 0u;
}

// ---------------------------------------------------------------------------
// K1: per-token softmax + top2. One thread per token (row of 64 f32).
//     Also deterministic integer expert histograms (pre-capacity counts).
// ---------------------------------------------------------------------------
__global__ void k1_softmax_top2(const float* __restrict__ in,
                                int* __restrict__ wsI, float* __restrict__ wsF) {
    __shared__ int h1[64], h2[64];
    if (threadIdx.x < 64) { h1[threadIdx.x] = 0; h2[threadIdx.x] = 0; }
    __syncthreads();

    int t = blockIdx.x * blockDim.x + threadIdx.x;
    float v[64];
    const v4f* row = (const v4f*)(in + (size_t)t * 64);
    #pragma unroll
    for (int i = 0; i < 16; i++) {
        v4f x = row[i];
        v[4*i] = x.x; v[4*i+1] = x.y; v[4*i+2] = x.z; v[4*i+3] = x.w;
    }
    // top-2 on logits (softmax is monotonic); strict '>' keeps lower index on ties
    float m1 = -INFINITY, m2 = -INFINITY; int i1 = 0, i2 = 0;
    #pragma unroll
    for (int e = 0; e < 64; e++) {
        float x = v[e];
        if (x > m1)      { m2 = m1; i2 = i1; m1 = x; i1 = e; }
        else if (x > m2) { m2 = x; i2 = e; }
    }
    float sum = 0.f;
    #pragma unroll
    for (int e = 0; e < 64; e++) sum += expf(v[e] - m1);
    float inv = 1.0f / sum;

    wsI[OFF_IDX1 + t] = i1;
    wsI[OFF_IDX2 + t] = i2;
    wsF[OFF_W1 + t]   = inv;                  // expf(m1-m1)*inv
    wsF[OFF_W2 + t]   = expf(m2 - m1) * inv;
    wsF[OFF_MX + t]   = m1;
    wsF[OFF_INV + t]  = inv;

    atomicAdd(&h1[i1], 1);
    atomicAdd(&h2[i2], 1);
    __syncthreads();
    if (threadIdx.x < 64) {
        atomicAdd(&wsI[OFF_CNT1 + threadIdx.x], h1[threadIdx.x]);
        atomicAdd(&wsI[OFF_CNT2 + threadIdx.x], h2[threadIdx.x]);
    }
}

// ---------------------------------------------------------------------------
// K2: me[e] = sum_t softmax(in[t])[e]  (deterministic tree reduction per expert)
// ---------------------------------------------------------------------------
__global__ void k2_me(const float* __restrict__ in, const float* __restrict__ wsF,
                      float* __restrict__ me) {
    int e = blockIdx.x;
    __shared__ float red[256];
    float p = 0.f;
    for (int t = threadIdx.x; t < S_TOKENS; t += 256)
        p += expf(in[(size_t)t * 64 + e] - wsF[OFF_MX + t]) * wsF[OFF_INV + t];
    red[threadIdx.x] = p;
    __syncthreads();
    for (int s = 128; s > 0; s >>= 1) {
        if (threadIdx.x < s) red[threadIdx.x] += red[threadIdx.x + s];
        __syncthreads();
    }
    if (threadIdx.x == 0) me[e] = red[0];
}

// ---------------------------------------------------------------------------
// K3: token-ordered cumsum of one-hot masks via WMMA matrix-scan + scatter.
//     Single block (256 thr = 8 wave32). Each wave owns a 16-token chunk per
//     iteration. Inclusive within-chunk cumsum of mask(16x64) is computed as
//     L16 x mask with v_wmma_f32_16x16x4_f32 (4 N-tiles x 4 K-steps).
//     Cross-wave/cross-chunk offsets via LDS totals + running counters.
// ---------------------------------------------------------------------------
__global__ void k3_scan_scatter(const int* __restrict__ wsI, const float* __restrict__ wsF,
                                float* __restrict__ combine, float* __restrict__ secmask) {
    __shared__ float cum[8 * 1024];   // per wave: 16 tokens x 64 experts
    __shared__ float wtot[8 * 64];    // per wave chunk totals
    __shared__ int run1[64], run2[64];

    int tid = threadIdx.x;
    if (tid < 64) { run1[tid] = 0; run2[tid] = 0; }
    __syncthreads();

    int lane = tid & 31;
    int w    = tid >> 5;
    bool hi  = lane >= 16;
    int M    = lane & 15;   // A-matrix row / B-matrix column / D-matrix column

    for (int iter = 0; iter < S_TOKENS / 128; iter++) {
        int base = iter * 128;
        for (int m = 0; m < 2; m++) {
            const int*   idxArr = wsI + (m == 0 ? OFF_IDX1 : OFF_IDX2);
            const float* wArr   = wsF + (m == 0 ? OFF_W1   : OFF_W2);
            int* run = (m == 0) ? run1 : run2;

            // each lane mirrors token (base + w*16 + M)'s expert for shfl broadcast
            int myIdx = idxArr[base + w * 16 + M];

            #pragma unroll
            for (int tile = 0; tile < 4; tile++) {
                v8f acc = {0.f, 0.f, 0.f, 0.f, 0.f, 0.f, 0.f, 0.f};
                #pragma unroll
                for (int k = 0; k < 4; k++) {
                    // 32-bit A/B frag layout: lo lanes hold K=4k+0,4k+1; hi: 4k+2,4k+3
                    int r0 = 4 * k + (hi ? 2 : 0);
                    int e0 = __shfl(myIdx, r0, 32);
                    int e1 = __shfl(myIdx, r0 + 1, 32);
                    int n  = tile * 16 + M;
                    v2f a, b;
                    a.x = (r0     <= M) ? 1.f : 0.f;   // inclusive lower-triangular L16
                    a.y = (r0 + 1 <= M) ? 1.f : 0.f;
                    b.x = (e0 == n) ? 1.f : 0.f;       // one-hot mask rows
                    b.y = (e1 == n) ? 1.f : 0.f;
                    acc = __builtin_amdgcn_wmma_f32_16x16x4_f32(
                        false, a, false, b, (short)0, acc, false, false);
                }
                // D layout: lanes 0-15 -> N=lane, rows M=vv; lanes 16-31 -> rows 8+vv
                int mb = hi ? 8 : 0;
                #pragma unroll
                for (int vv = 0; vv < 8; vv++)
                    cum[w * 1024 + (mb + vv) * 64 + tile * 16 + M] = acc[vv];
                if (hi) wtot[w * 64 + tile * 16 + M] = acc[7];  // inclusive row 15 = total
            }
            __syncthreads();

            if (tid < 128) {
                int t   = base + tid;
                int wt  = tid >> 4;
                int row = tid & 15;
                int e   = idxArr[t];
                // exclusive rank = running + inclusive_within_chunk - 1 + earlier waves
                int rank = run[e] + (int)cum[wt * 1024 + row * 64 + e] - 1;
                for (int ww = 0; ww < wt; ww++) rank += (int)wtot[ww * 64 + e];
                if (m == 1) rank += wsI[OFF_CNT1 + e];  // locations2 += sum(mask1)
                if (rank < CAPACITY) {
                    size_t off = ((size_t)t * E_EXPERTS + e) * CAPACITY + rank;
                    combine[off] = wArr[t];
                    secmask[off] = 1.0f;
                }
            }
            __syncthreads();

            if (tid < 64) {
                int add = 0;
                #pragma unroll
                for (int ww = 0; ww < 8; ww++) add += (int)wtot[ww * 64 + tid];
                run[tid] += add;
            }
            __syncthreads();
        }
    }
}

// ---------------------------------------------------------------------------
// K4: exp_counts (pre-capacity) and l_aux = E * sum(me*ce) / 2
// ---------------------------------------------------------------------------
__global__ void k4_finalize(const int* __restrict__ wsI, const float* __restrict__ wsF,
                            float* __restrict__ out_tail) {
    __shared__ float red[64];
    int e = threadIdx.x; // 64 threads
    int c1 = wsI[OFF_CNT1 + e], c2 = wsI[OFF_CNT2 + e];
    float ec = (float)(c1 + c2);
    out_tail[e] = ec;
    float ce  = ec / (float)S_TOKENS;
    float mem = wsF[OFF_ME + e] / (float)S_TOKENS;
    red[e] = mem * ce;
    __syncthreads();
    for (int s = 32; s > 0; s >>= 1) {
        if (e < s) red[e] += red[e + s];
        __syncthreads();
    }
    if (e == 0) out_tail[64] = (float)E_EXPERTS * red[0] * 0.5f;
}

extern "C" void kernel_launch(void* const* d_in, const int* in_sizes, int n_in,
                              void* d_out, int out_size, void* d_ws, size_t ws_size,
                              hipStream_t stream) {
    const float* in = (const float*)d_in[0];
    float* out = (float*)d_out;
    int*      wsI = (int*)d_ws;
    float*    wsF = (float*)d_ws;
    unsigned* wsw = (unsigned*)d_ws;

    float* combine = out;
    float* secmask = out + SEC;
    float* tail    = out + 2 * SEC;

    k0_zero<<<8192, 256, 0, stream>>>(out, wsw);
    k1_softmax_top2<<<S_TOKENS / 256, 256, 0, stream>>>(in, wsI, wsF);
    k2_me<<<64, 256, 0, stream>>>(in, wsF, wsF + OFF_ME);
    k3_scan_scatter<<<1, 256, 0, stream>>>(wsI, wsF, combine, secmask);
    k4_finalize<<<1, 64, 0, stream>>>(wsI, wsF, tail);
}